// GBST_55662776156188
// MI455X (gfx1250) — compile-verified
//
#include <hip/hip_runtime.h>

typedef __attribute__((ext_vector_type(2))) float v2f;
typedef __attribute__((ext_vector_type(8))) float v8f;

#define BATCH 8
#define DIN   16
#define LL    4096
#define DD    512
#define NP    4104          // ceil(4096/12)*12
#define NTILES 342          // NP/12
#define NG    1024          // L/DS

// workspace layout (in floats)
#define H_OFF  0ULL
#define S_OFF  16777216ULL                 // B*L*D
#define SN_OFF (S_OFF + (unsigned long long)BATCH * NP * 4ULL)

// ---------------------------------------------------------------------------
// Phase 1: conv as GEMM  h[(b,n), d] = sum_{ik<64} X[(b,n),ik] * W[ik,d] + bias
// One wave computes one 16x16 tile of h via 16 x V_WMMA_F32_16X16X4_F32.
// A-fragment window offsets are k-step invariant: lane-half 0 reads cols
// {n-1, n}, half 1 reads {n+1, n+2}; masks/clamps hoisted -> branchless loop.
// B-fragment: two consecutive floats at an 8B-aligned address -> one b64 load.
// ---------------------------------------------------------------------------
__global__ __launch_bounds__(256) void gbst_conv_wmma(
    const float* __restrict__ x, const float* __restrict__ w,
    const float* __restrict__ bias, float* __restrict__ h)
{
  const int wave = threadIdx.x >> 5;
  const int lane = threadIdx.x & 31;
  const int mtile = blockIdx.x * 8 + wave;     // 0..2047  (M = 32768 rows / 16)
  const int d0 = blockIdx.y * 16;              // N tile
  const int m0 = mtile * 16;
  const int bat = m0 >> 12;                    // L = 4096 rows per batch, tile never straddles
  const int nbase = m0 & 4095;

  // A fragment mapping (32-bit A 16x4): lanes 0-15 hold K={0,1}, lanes 16-31 K={2,3}
  const int mA = lane & 15;
  const int khalf = (lane >> 4) * 2;
  const int nrowA = nbase + mA;
  // B fragment mapping (4x16): lane&15 = column d, lane half selects K pair
  const int dB = d0 + (lane & 15);

  const float* xb = x + (size_t)bat * DIN * LL;
  const float* wb = w + (size_t)dB * 64;       // conv_w[d][i][k], ik = i*4+k

  // Hoisted, k-invariant padding logic (padding (1,2): window n-1..n+2)
  const int base  = nrowA + khalf - 1;         // half0: n-1, half1: n+1
  const bool ok0  = (base >= 0) && (base < LL);
  const bool ok1  = (base + 1 < LL);           // base+1 >= 0 always
  const int  c0   = ok0 ? base : 0;
  const int  c1   = ok1 ? (base + 1) : 0;
  const float msk0 = ok0 ? 1.f : 0.f;
  const float msk1 = ok1 ? 1.f : 0.f;

  v8f c = {0.f, 0.f, 0.f, 0.f, 0.f, 0.f, 0.f, 0.f};

  #pragma unroll
  for (int kk = 0; kk < 16; ++kk) {            // i = kk, K=4 per WMMA
    const float* xr = xb + kk * LL;
    v2f a;
    a.x = xr[c0] * msk0;
    a.y = xr[c1] * msk1;
    const v2f bfrag = *(const v2f*)(wb + kk * 4 + khalf);  // 8B aligned
    c = __builtin_amdgcn_wmma_f32_16x16x4_f32(false, a, false, bfrag,
                                              (short)0, c, false, false);
  }

  // C/D layout: reg r, lanes 0-15 -> M=r ; lanes 16-31 -> M=r+8 ; N = lane&15
  const float bb = bias[dB];
  const int mhalf = (lane >> 4) * 8;
  #pragma unroll
  for (int r = 0; r < 8; ++r) {
    const int nrow = nbase + r + mhalf;
    h[(((size_t)bat * LL) + nrow) * DD + dB] = c[r] + bb;
  }
}

// ---------------------------------------------------------------------------
// Phase 2: s[b,n] = dot(h[b,n,:], score_w); block means m=1..4 within aligned
// 12-row tiles (lcm(1,2,3,4)=12=PAD_MULT); softmax over m -> S (B, NP, 4).
// Rows >= L are zero-padded (reference pads h with zeros after bias).
// ---------------------------------------------------------------------------
__global__ __launch_bounds__(384) void gbst_score(
    const float* __restrict__ h, const float* __restrict__ score_w,
    const float* __restrict__ score_b, float* __restrict__ Sout)
{
  __shared__ float part[384];
  __shared__ float srow[12];
  const int tid  = threadIdx.x;
  const int wave = tid >> 5;
  const int lane = tid & 31;
  const int tile = blockIdx.x % NTILES;
  const int bat  = blockIdx.x / NTILES;
  const int n    = tile * 12 + wave;           // wave w owns row w of the tile

  float acc = 0.f;
  if (n < LL) {
    const float* hp = h + (((size_t)bat * LL) + n) * DD;
    #pragma unroll
    for (int it = 0; it < 16; ++it)
      acc += hp[lane + it * 32] * score_w[lane + it * 32];
  }
  part[tid] = acc;
  __syncthreads();

  if (tid < 12) {
    float s = 0.f;
    #pragma unroll
    for (int l = 0; l < 32; ++l) s += part[tid * 32 + l];
    srow[tid] = s;
  }
  __syncthreads();

  if (tid < 12) {
    const int rr = tid;
    const float m1 = srow[rr];
    const int b2 = rr & ~1;
    const float m2 = 0.5f * (srow[b2] + srow[b2 + 1]);
    const int b3 = (rr / 3) * 3;
    const float m3 = (srow[b3] + srow[b3 + 1] + srow[b3 + 2]) * (1.f / 3.f);
    const int b4 = rr & ~3;
    const float m4 = 0.25f * (srow[b4] + srow[b4 + 1] + srow[b4 + 2] + srow[b4 + 3]);
    const float sb = score_b[0];
    const float v0 = m1 + sb, v1 = m2 + sb, v2 = m3 + sb, v3 = m4 + sb;
    const float mx = fmaxf(fmaxf(v0, v1), fmaxf(v2, v3));
    const float e0 = __expf(v0 - mx), e1 = __expf(v1 - mx);
    const float e2 = __expf(v2 - mx), e3 = __expf(v3 - mx);
    const float inv = 1.f / (e0 + e1 + e2 + e3);
    float4* o = (float4*)(Sout + (((size_t)bat * NP) + tile * 12 + rr) * 4);
    *o = make_float4(e0 * inv, e1 * inv, e2 * inv, e3 * inv);
  }
}

// ---------------------------------------------------------------------------
// Phase 3: attention over positions, head dim 4.  S (4104x4) lives in LDS
// (65,664 B of the 320 KB WGP LDS).  sim = qk in [0,1] because S rows are
// probability vectors -> exp cannot overflow -> single-pass softmax, no max.
// ---------------------------------------------------------------------------
__global__ __launch_bounds__(256) void gbst_attn(
    const float* __restrict__ Sin, float* __restrict__ Sout)
{
  extern __shared__ float4 sS[];               // NP float4 = 65,664 bytes
  const int bat = blockIdx.y;
  const float4* src = (const float4*)(Sin + (size_t)bat * NP * 4);
  for (int idx = threadIdx.x; idx < NP; idx += 256) sS[idx] = src[idx];
  __syncthreads();

  const int i = blockIdx.x * 256 + threadIdx.x;
  if (i >= NP) return;
  const float4 q = sS[i];
  float sum = 0.f;
  float ox = 0.f, oy = 0.f, oz = 0.f, ow = 0.f;
  #pragma unroll 4
  for (int j = 0; j < NP; ++j) {
    const float4 kj = sS[j];
    const float d = q.x * kj.x + q.y * kj.y + q.z * kj.z + q.w * kj.w;
    const float e = __expf(d);                 // d in [0,1]: safe
    sum += e;
    ox += e * kj.x; oy += e * kj.y; oz += e * kj.z; ow += e * kj.w;
  }
  const float inv = 1.f / sum;
  float4* dst = (float4*)(Sout + (((size_t)bat * NP) + i) * 4);
  *dst = make_float4(ox * inv, oy * inv, oz * inv, ow * inv);
}

// ---------------------------------------------------------------------------
// Phase 4: xo[b,n,d] = sum_m S'[b,n,m] * blockmean_m(h)[b,n,d]; 12-row tiles
// are self-contained for all block sizes.  Also emit the DS=4 down-sample.
// ---------------------------------------------------------------------------
__global__ __launch_bounds__(256) void gbst_output(
    const float* __restrict__ h, const float* __restrict__ Sn,
    float* __restrict__ out1, float* __restrict__ out2)
{
  __shared__ float4 wgt[12];
  const int tile = blockIdx.x;
  const int bat  = blockIdx.y;
  const int n0   = tile * 12;
  if (threadIdx.x < 12)
    wgt[threadIdx.x] = *(const float4*)(Sn + (((size_t)bat * NP) + n0 + threadIdx.x) * 4);
  __syncthreads();

  #pragma unroll
  for (int cc = 0; cc < 2; ++cc) {
    const int d = threadIdx.x + cc * 256;      // coalesced: adjacent lanes -> adjacent d
    float hv[12];
    #pragma unroll
    for (int r = 0; r < 12; ++r) {
      const int n = n0 + r;
      hv[r] = (n < LL) ? h[(((size_t)bat * LL) + n) * DD + d] : 0.f;
    }
    float p2[6];
    #pragma unroll
    for (int r = 0; r < 6; ++r) p2[r] = hv[2 * r] + hv[2 * r + 1];
    float t3[4];
    #pragma unroll
    for (int r = 0; r < 4; ++r) t3[r] = hv[3 * r] + hv[3 * r + 1] + hv[3 * r + 2];
    float q4[3];
    #pragma unroll
    for (int r = 0; r < 3; ++r) q4[r] = p2[2 * r] + p2[2 * r + 1];

    float xo[12];
    #pragma unroll
    for (int r = 0; r < 12; ++r) {
      const float4 wv = wgt[r];
      xo[r] = wv.x * hv[r]
            + wv.y * (0.5f * p2[r >> 1])
            + wv.z * ((1.f / 3.f) * t3[r / 3])
            + wv.w * (0.25f * q4[r >> 2]);
      const int n = n0 + r;
      if (n < LL) out1[(((size_t)bat * LL) + n) * DD + d] = xo[r];
    }
    #pragma unroll
    for (int g = 0; g < 3; ++g) {
      const int grp = tile * 3 + g;
      if (grp < NG)
        out2[(((size_t)bat * NG) + grp) * DD + d] =
            0.25f * (xo[4 * g] + xo[4 * g + 1] + xo[4 * g + 2] + xo[4 * g + 3]);
    }
  }
}

// ---------------------------------------------------------------------------
extern "C" void kernel_launch(void* const* d_in, const int* in_sizes, int n_in,
                              void* d_out, int out_size, void* d_ws, size_t ws_size,
                              hipStream_t stream) {
  (void)in_sizes; (void)n_in; (void)out_size; (void)ws_size;
  const float* x       = (const float*)d_in[0];
  const float* conv_w  = (const float*)d_in[1];
  const float* conv_b  = (const float*)d_in[2];
  const float* score_w = (const float*)d_in[3];
  const float* score_b = (const float*)d_in[4];
  float* out = (float*)d_out;
  float* ws  = (float*)d_ws;

  float* h  = ws + H_OFF;    // B*L*D floats (64 MB, fits in 192 MB L2)
  float* S  = ws + S_OFF;    // B*NP*4
  float* Sn = ws + SN_OFF;   // B*NP*4

  // Phase 1: 2048 M-tiles x 32 N-tiles, 8 waves (M-tiles) per block
  gbst_conv_wmma<<<dim3(256, 32), 256, 0, stream>>>(x, conv_w, conv_b, h);
  // Phase 2: one 12-row tile per block, 12 waves
  gbst_score<<<dim3(BATCH * NTILES), 384, 0, stream>>>(h, score_w, score_b, S);
  // Phase 3: 17 row-tiles x 8 batches, S in dynamic LDS
  gbst_attn<<<dim3((NP + 255) / 256, BATCH), 256, NP * sizeof(float4), stream>>>(S, Sn);
  // Phase 4: 342 tiles x 8 batches
  gbst_output<<<dim3(NTILES, BATCH), 256, 0, stream>>>(h, Sn, out,
                                                       out + (size_t)BATCH * LL * DD);
}